// Gemma3nTextAltUp_27719718928500
// MI455X (gfx1250) — compile-verified
//
#include <hip/hip_runtime.h>
#include <math.h>

// ---- problem constants (from reference) ----
#define HIDDEN  2048
#define NUMIN   4
#define NTOK    (4 * 2048)        // B*S tokens
#define EPSV    1e-6f

typedef __attribute__((ext_vector_type(2))) float v2f;
typedef __attribute__((ext_vector_type(8))) float v8f;

#if defined(__has_builtin)
#if __has_builtin(__builtin_amdgcn_wmma_f32_16x16x4_f32)
#define USE_WMMA_F32_16X16X4 1
#endif
#endif

// =====================================================================
// Kernel 1: router / compute_modalities via V_WMMA_F32_16X16X4_F32.
// One wave (32 lanes) handles a 16-token tile; blockIdx.y selects the
// input (0: hidden_states[ACTIVE_IDX=0] -> mod_p, 1: activated -> mod_c).
//
// A-matrix (16 tokens x 4 K) ISA layout: lane L holds token L%16,
//   K = vgpr + 2*(L>=16) relative to the K-slice base.
// B-matrix (4 K x 16 cols): col = L%16 (cols >= 4 zero-masked),
//   K = vgpr + 2*(L>=16).
// C/D 16x16 f32: VGPR r, lane L -> row (token) = r + 8*(L>=16), col = L%16.
// Sum-of-squares for RMSNorm rides on the same loads: lanes L and L+16
// jointly cover all K of token L%16, so one shfl_xor(16) completes it.
// =====================================================================
__global__ __launch_bounds__(32)
void router_wmma_kernel(const float* __restrict__ hs0,
                        const float* __restrict__ act,
                        const float* __restrict__ norm_scale,
                        const float* __restrict__ router_w,   // [H][4]
                        float* __restrict__ mod_out)          // [2][NTOK][4]
{
    const int tile  = blockIdx.x;          // 16 tokens per tile
    const int which = blockIdx.y;          // 0 = predict, 1 = correct
    const float* x_base = which ? act : hs0;
    float* out = mod_out + (size_t)which * NTOK * NUMIN;

    const int lane  = threadIdx.x;
    const int trow  = lane & 15;           // token-in-tile (A rows / C cols)
    const int half  = lane >> 4;
    const int token = tile * 16 + trow;
    const float* x  = x_base + (size_t)token * HIDDEN;

    const int   ncol  = trow & 3;                          // clamped B column
    const float nmask = (trow < NUMIN) ? 1.0f : 0.0f;      // zero cols 4..15

    v8f acc0 = {};
    v8f acc1 = {};
    float sumsq = 0.0f;

    for (int k0 = 0; k0 < HIDDEN; k0 += 8) {
        // ---- WMMA #0 : K = k0 .. k0+3 ----
        const int ka = k0 + half * 2;
        v2f xa = *(const v2f*)(x + ka);
        sumsq += xa.x * xa.x + xa.y * xa.y;
        v2f sa = *(const v2f*)(norm_scale + ka);
        v2f A0; A0.x = xa.x * sa.x; A0.y = xa.y * sa.y;
        v2f B0;
        B0.x = router_w[(ka + 0) * NUMIN + ncol] * nmask;
        B0.y = router_w[(ka + 1) * NUMIN + ncol] * nmask;
#ifdef USE_WMMA_F32_16X16X4
        acc0 = __builtin_amdgcn_wmma_f32_16x16x4_f32(false, A0, false, B0,
                                                     (short)0, acc0,
                                                     false, false);
#else
        acc0[0] += A0.x * B0.x + A0.y * B0.y;   // placeholder fallback
#endif

        // ---- WMMA #1 : K = k0+4 .. k0+7 (independent accumulator) ----
        const int kb = ka + 4;
        v2f xb = *(const v2f*)(x + kb);
        sumsq += xb.x * xb.x + xb.y * xb.y;
        v2f sb = *(const v2f*)(norm_scale + kb);
        v2f A1; A1.x = xb.x * sb.x; A1.y = xb.y * sb.y;
        v2f B1;
        B1.x = router_w[(kb + 0) * NUMIN + ncol] * nmask;
        B1.y = router_w[(kb + 1) * NUMIN + ncol] * nmask;
#ifdef USE_WMMA_F32_16X16X4
        acc1 = __builtin_amdgcn_wmma_f32_16x16x4_f32(false, A1, false, B1,
                                                     (short)0, acc1,
                                                     false, false);
#else
        acc1[0] += A1.x * B1.x + A1.y * B1.y;
#endif
    }

    v8f acc;
#pragma unroll
    for (int i = 0; i < 8; ++i) acc[i] = acc0[i] + acc1[i];

    // finish RMSNorm reduction: pair (L, L+16) covers all K of token L%16
    sumsq += __shfl_xor(sumsq, 16, 32);
    const float inv = rsqrtf(sumsq * (1.0f / HIDDEN) + EPSV);

#pragma unroll
    for (int r = 0; r < 8; ++r) {
        const int tt   = r + 8 * half;            // token-in-tile of acc[r]
        const float it = __shfl(inv, tt, 32);     // inv lives at lane tt
        const float routed = acc[r] * it * (1.0f / HIDDEN);
        const float mval   = tanhf(routed);
        if (trow < NUMIN)
            out[(size_t)(tile * 16 + tt) * NUMIN + trow] = mval;
    }
}

// =====================================================================
// Kernel 2: tiny per-token matmuls (4x16 and 4x4) -> raw[16], coefs[4].
// =====================================================================
__global__ __launch_bounds__(256)
void token_coef_kernel(const float* __restrict__ mod,     // [2][NTOK][4]
                       const float* __restrict__ pred_w,  // [4][16]
                       const float* __restrict__ corr_w,  // [4][4]
                       float* __restrict__ raw_out,       // [NTOK][16]
                       float* __restrict__ coef_out)      // [NTOK][4]
{
    const int t = blockIdx.x * blockDim.x + threadIdx.x;
    if (t >= NTOK) return;

    float mp[4], mc[4];
#pragma unroll
    for (int j = 0; j < 4; ++j) {
        mp[j] = mod[(size_t)t * 4 + j];
        mc[j] = mod[(size_t)NTOK * 4 + (size_t)t * 4 + j];
    }
#pragma unroll
    for (int k = 0; k < 16; ++k) {
        float s = 0.0f;
#pragma unroll
        for (int j = 0; j < 4; ++j) s += mp[j] * pred_w[j * 16 + k];
        raw_out[(size_t)t * 16 + k] = s;
    }
#pragma unroll
    for (int m = 0; m < 4; ++m) {
        float s = 1.0f;
#pragma unroll
        for (int j = 0; j < 4; ++j) s += mc[j] * corr_w[j * 4 + m];
        coef_out[(size_t)t * 4 + m] = s;
    }
}

// =====================================================================
// Kernel 3: fused predict + correct + output-scale, the bandwidth pass.
// One thread = 4 consecutive h for one token, all 4 output modalities.
// Streams 4x float4 hs loads + act + writes 4x float4 -> pure BW bound.
// =====================================================================
__global__ __launch_bounds__(256)
void fuse_kernel(const float* __restrict__ hs,     // [4][NTOK][H]
                 const float* __restrict__ act,    // [NTOK][H]
                 const float* __restrict__ raw,    // [NTOK][16]
                 const float* __restrict__ coef,   // [NTOK][4]
                 const float* __restrict__ oscale, // [H]
                 float* __restrict__ out)          // [4][NTOK][H]
{
    const int HV = HIDDEN / 4;
    const size_t idx = (size_t)blockIdx.x * blockDim.x + threadIdx.x;
    const int t  = (int)(idx / HV);
    const int h4 = (int)(idx % HV) * 4;
    if (t >= NTOK) return;

    float r[16];
#pragma unroll
    for (int i = 0; i < 16; ++i) r[i] = raw[(size_t)t * 16 + i];
    float c[4];
#pragma unroll
    for (int m = 0; m < 4; ++m) c[m] = coef[(size_t)t * 4 + m];

    const size_t base = (size_t)t * HIDDEN + h4;
    float4 h[4];
#pragma unroll
    for (int n = 0; n < 4; ++n)
        h[n] = *(const float4*)(hs + (size_t)n * NTOK * HIDDEN + base);
    const float4 a  = *(const float4*)(act + base);
    const float4 os = *(const float4*)(oscale + h4);

    float4 pred[4];
#pragma unroll
    for (int m = 0; m < 4; ++m) {
        pred[m].x = h[m].x + r[m*4+0]*h[0].x + r[m*4+1]*h[1].x + r[m*4+2]*h[2].x + r[m*4+3]*h[3].x;
        pred[m].y = h[m].y + r[m*4+0]*h[0].y + r[m*4+1]*h[1].y + r[m*4+2]*h[2].y + r[m*4+3]*h[3].y;
        pred[m].z = h[m].z + r[m*4+0]*h[0].z + r[m*4+1]*h[1].z + r[m*4+2]*h[2].z + r[m*4+3]*h[3].z;
        pred[m].w = h[m].w + r[m*4+0]*h[0].w + r[m*4+1]*h[1].w + r[m*4+2]*h[2].w + r[m*4+3]*h[3].w;
    }
    float4 inn;
    inn.x = a.x - pred[0].x;
    inn.y = a.y - pred[0].y;
    inn.z = a.z - pred[0].z;
    inn.w = a.w - pred[0].w;

#pragma unroll
    for (int m = 0; m < 4; ++m) {
        float4 o;
        o.x = (pred[m].x + inn.x * c[m]) * os.x;
        o.y = (pred[m].y + inn.y * c[m]) * os.y;
        o.z = (pred[m].z + inn.z * c[m]) * os.z;
        o.w = (pred[m].w + inn.w * c[m]) * os.w;
        *(float4*)(out + (size_t)m * NTOK * HIDDEN + base) = o;
    }
}

// =====================================================================
extern "C" void kernel_launch(void* const* d_in, const int* in_sizes, int n_in,
                              void* d_out, int out_size, void* d_ws, size_t ws_size,
                              hipStream_t stream) {
    const float* hs     = (const float*)d_in[0]; // [4][B][S][H]; ACTIVE_IDX=0 slice is the base
    const float* act    = (const float*)d_in[1]; // [B][S][H]
    const float* nscale = (const float*)d_in[2]; // [H]
    const float* rw     = (const float*)d_in[3]; // [H][4]
    const float* pw     = (const float*)d_in[4]; // [4][16]
    const float* cw     = (const float*)d_in[5]; // [4][4]
    const float* osc    = (const float*)d_in[6]; // [H]
    float* out = (float*)d_out;

    // workspace layout (floats): mod[2*T*4] | raw[T*16] | coef[T*4]  (~0.9 MB)
    float* ws   = (float*)d_ws;
    float* mod  = ws;
    float* raw  = ws + (size_t)2 * NTOK * NUMIN;
    float* coef = raw + (size_t)NTOK * 16;

    dim3 g1(NTOK / 16, 2);
    router_wmma_kernel<<<g1, 32, 0, stream>>>(hs, act, nscale, rw, mod);

    token_coef_kernel<<<(NTOK + 255) / 256, 256, 0, stream>>>(mod, pw, cw, raw, coef);

    const size_t total = (size_t)NTOK * (HIDDEN / 4);
    fuse_kernel<<<(unsigned)((total + 255) / 256), 256, 0, stream>>>(
        hs, act, raw, coef, osc, out);
}